// GPLoss_83279415869695
// MI455X (gfx1250) — compile-verified
//
#include <hip/hip_runtime.h>

typedef __attribute__((ext_vector_type(2))) float v2f;
typedef __attribute__((ext_vector_type(8))) float v8f;

static constexpr int N   = 160;
static constexpr int SH_ = 160;            // stride along H
static constexpr int SD_ = 160 * 160;      // stride along D
static constexpr int SB_ = 160 * 160 * 160;// stride along B
#define EPS_N 1e-12f

// ---------------------------------------------------------------------------
// Wave32 full reduction via V_WMMA_F32_16X16X4_F32.
// A: VGPR0 = v (lane), VGPR1 = 0  =>  A[m][0]=v[m], A[m][2]=v[m+16], else 0.
// B: ones(4x16) (both VGPRs = 1.0 in every lane; layout-independent).
// D[m][n] = v[m] + v[m+16] for all n.  D-vgpr j holds r[j] in lanes 0-15 and
// r[j+8] in lanes 16-31, so summing the 8 D vgprs per-lane gives the two
// half-sums; one xor-16 shuffle yields the full 32-lane sum in every lane.
// ---------------------------------------------------------------------------
__device__ __forceinline__ float wave_sum32(float v) {
    v2f a; a.x = v;    a.y = 0.0f;
    v2f b; b.x = 1.0f; b.y = 1.0f;
    v8f c = {};
    c = __builtin_amdgcn_wmma_f32_16x16x4_f32(false, a, false, b,
                                              (short)0, c, false, false);
    float s = c[0] + c[1] + c[2] + c[3] + c[4] + c[5] + c[6] + c[7];
    s += __shfl_xor(s, 16, 32);
    return s;
}

__device__ __forceinline__ float cos_of(float sft, float sff, float stt) {
    return sft / (fmaxf(sqrtf(sff), EPS_N) * fmaxf(sqrtf(stt), EPS_N));
}

// ---------------------------------------------------------------------------
// Strided lines (line axis = H or D). One thread owns a whole line; lanes run
// across consecutive w -> coalesced. Computes BOTH gradient combos sharing
// this line orientation: grad along the cross axis (pA) and grad along W (pB).
// Boundary lines use stride 0 -> zero gradient -> cos contribution 0.
// ---------------------------------------------------------------------------
template <int S_LINE, int S_CROSS>
__global__ __launch_bounds__(256) void sploss_strided(const float* __restrict__ f,
                                                      const float* __restrict__ t,
                                                      float* __restrict__ pA,
                                                      float* __restrict__ pB) {
    const int li = blockIdx.x * 256 + threadIdx.x;   // 51200 lines total
    const int w  = li % N;
    const int c  = (li / N) % N;
    const int b  = li / (N * N);
    const int base = b * SB_ + c * S_CROSS + w;
    const int sgc = (c < N - 1) ? S_CROSS : 0;       // grad along cross axis
    const int sgw = (w < N - 1) ? 1 : 0;             // grad along W

    float aft = 0.f, aff = 0.f, att = 0.f;
    float bft = 0.f, bff = 0.f, btt = 0.f;
#pragma unroll 4
    for (int i = 0; i < N; ++i) {
        const int p = base + i * S_LINE;
        const float f0 = f[p], t0 = t[p];
        const float gfc = f[p + sgc] - f0, gtc = t[p + sgc] - t0;
        const float gfw = f[p + sgw] - f0, gtw = t[p + sgw] - t0;
        aft = fmaf(gfc, gtc, aft); aff = fmaf(gfc, gfc, aff); att = fmaf(gtc, gtc, att);
        bft = fmaf(gfw, gtw, bft); bff = fmaf(gfw, gfw, bff); btt = fmaf(gtw, gtw, btt);
    }
    const float cosA = cos_of(aft, aff, att);
    const float cosB = cos_of(bft, bff, btt);

    __shared__ float redA[256];
    __shared__ float redB[256];
    redA[threadIdx.x] = cosA;
    redB[threadIdx.x] = cosB;
    __syncthreads();
    for (int s = 128; s > 0; s >>= 1) {
        if (threadIdx.x < s) {
            redA[threadIdx.x] += redA[threadIdx.x + s];
            redB[threadIdx.x] += redB[threadIdx.x + s];
        }
        __syncthreads();
    }
    if (threadIdx.x == 0) { pA[blockIdx.x] = redA[0]; pB[blockIdx.x] = redB[0]; }
}

// ---------------------------------------------------------------------------
// Contiguous lines (line axis = W). One wave per line (lane covers w=lane+32k,
// coalesced); WMMA-based wave reductions; computes combos grad-D and grad-H.
// All waves fully active -> EXEC all-1s at the WMMA as required.
// ---------------------------------------------------------------------------
__global__ __launch_bounds__(256) void sploss_wlines(const float* __restrict__ f,
                                                     const float* __restrict__ t,
                                                     float* __restrict__ pD,
                                                     float* __restrict__ pH) {
    const int lane   = threadIdx.x & 31;
    const int wv     = threadIdx.x >> 5;
    const int waveId = blockIdx.x * 8 + wv;          // 3200 waves, 16 lines each
    float locD = 0.f, locH = 0.f;

    for (int j = 0; j < 16; ++j) {
        const int li = waveId * 16 + j;
        const int h  = li % N;
        const int d  = (li / N) % N;
        const int b  = li / (N * N);
        const int base = b * SB_ + d * SD_ + h * SH_;
        const int sgd  = (d < N - 1) ? SD_ : 0;
        const int sgh  = (h < N - 1) ? SH_ : 0;

        float dft = 0.f, dff = 0.f, dtt = 0.f;
        float hft = 0.f, hff = 0.f, htt = 0.f;
#pragma unroll
        for (int k = 0; k < 5; ++k) {
            const int p = base + lane + 32 * k;
            const float f0 = f[p], t0 = t[p];
            const float gfd = f[p + sgd] - f0, gtd = t[p + sgd] - t0;
            const float gfh = f[p + sgh] - f0, gth = t[p + sgh] - t0;
            dft = fmaf(gfd, gtd, dft); dff = fmaf(gfd, gfd, dff); dtt = fmaf(gtd, gtd, dtt);
            hft = fmaf(gfh, gth, hft); hff = fmaf(gfh, gfh, hff); htt = fmaf(gth, gth, htt);
        }
        const float SftD = wave_sum32(dft);
        const float SffD = wave_sum32(dff);
        const float SttD = wave_sum32(dtt);
        const float SftH = wave_sum32(hft);
        const float SffH = wave_sum32(hff);
        const float SttH = wave_sum32(htt);
        if (lane == 0) {
            locD += cos_of(SftD, SffD, SttD);
            locH += cos_of(SftH, SffH, SttH);
        }
    }

    __shared__ float rD[8];
    __shared__ float rH[8];
    if (lane == 0) { rD[wv] = locD; rH[wv] = locH; }
    __syncthreads();
    if (threadIdx.x == 0) {
        float sD = 0.f, sH = 0.f;
        for (int i = 0; i < 8; ++i) { sD += rD[i]; sH += rH[i]; }
        pD[blockIdx.x] = sD;
        pH[blockIdx.x] = sH;
    }
}

// ---------------------------------------------------------------------------
// Deterministic final combine: fixed slot ranges, fixed tree order.
// Regions r=0..5 at offset r*512: r<4 have 200 partials, r>=4 have 400.
// ---------------------------------------------------------------------------
__global__ __launch_bounds__(256) void final_combine(const float* __restrict__ part,
                                                     float* __restrict__ out) {
    const int tid = threadIdx.x;
    float s = 0.f;
#pragma unroll
    for (int r = 0; r < 6; ++r) {
        const int off = r * 512;
        const int cnt = (r < 4) ? 200 : 400;
        for (int i = tid; i < cnt; i += 256) s += part[off + i];
    }
    __shared__ float red[256];
    red[tid] = s;
    __syncthreads();
    for (int k = 128; k > 0; k >>= 1) {
        if (tid < k) red[tid] += red[tid + k];
        __syncthreads();
    }
    if (tid == 0) out[0] = -red[0] / 960.0f;   // all six cos-sums / (320*3), negated
}

extern "C" void kernel_launch(void* const* d_in, const int* in_sizes, int n_in,
                              void* d_out, int out_size, void* d_ws, size_t ws_size,
                              hipStream_t stream) {
    (void)in_sizes; (void)n_in; (void)out_size; (void)ws_size;
    const float* fake = (const float*)d_in[0];
    const float* tru  = (const float*)d_in[1];
    float* part = (float*)d_ws;   // 6 regions * 512 floats = 12 KB
    float* out  = (float*)d_out;

    // L = H lines: combos (grad D) -> region 0, (grad W) -> region 1
    sploss_strided<SH_, SD_><<<200, 256, 0, stream>>>(fake, tru, part + 0 * 512, part + 1 * 512);
    // L = D lines: combos (grad H) -> region 2, (grad W) -> region 3
    sploss_strided<SD_, SH_><<<200, 256, 0, stream>>>(fake, tru, part + 2 * 512, part + 3 * 512);
    // L = W lines: combos (grad D) -> region 4, (grad H) -> region 5
    sploss_wlines<<<400, 256, 0, stream>>>(fake, tru, part + 4 * 512, part + 5 * 512);

    final_combine<<<1, 256, 0, stream>>>(part, out);
}